// SSD_loss_3762391351637
// MI455X (gfx1250) — compile-verified
//
#include <hip/hip_runtime.h>
#include <math.h>

#define ALPHA_C  0.5f
#define GAMMA_C  1.5f
#define THRESH_C 0.5f
#define EPS8_C   1e-8f
#define EPS7_C   1e-7f

typedef __attribute__((ext_vector_type(2))) float v2f;
typedef __attribute__((ext_vector_type(8))) float v8f;
typedef __attribute__((ext_vector_type(4))) unsigned int u32x4;
typedef __attribute__((ext_vector_type(4))) int i32x4;
typedef __attribute__((ext_vector_type(8))) int i32x8;

// ---------------------------------------------------------------------------
// Wave-level sum reduction on the matrix pipe: D = A(16x4) * ones(4x16).
// a = {val, 0} -> row m of A = [val_m, 0, val_{m+16}, 0]; every D element on
// a lane is a row-sum pair; summing the 8 D VGPRs + one xor-16 shuffle gives
// the full 32-lane sum in every lane. EXEC all-ones at call sites.
// ---------------------------------------------------------------------------
__device__ __forceinline__ float wave_sum_wmma(float v) {
  v2f a; a.x = v;   a.y = 0.0f;
  v2f b; b.x = 1.0f; b.y = 1.0f;
  v8f c = {};
  c = __builtin_amdgcn_wmma_f32_16x16x4_f32(false, a, false, b, (short)0, c,
                                            false, false);
  float s = c[0] + c[1] + c[2] + c[3] + c[4] + c[5] + c[6] + c[7];
  s += __shfl_xor(s, 16, 32);
  return s;
}

// Block (256 threads = 8 waves) reduction -> one global atomicAdd.
__device__ __forceinline__ void block_accum(float v, float* dst, float* red8) {
  float w = wave_sum_wmma(v);
  const int lane = threadIdx.x & 31;
  const int wid  = threadIdx.x >> 5;
  if (lane == 0) red8[wid] = w;
  __syncthreads();
  if (threadIdx.x == 0) {
    float s = 0.f;
#pragma unroll
    for (int i = 0; i < 8; ++i) s += red8[i];
    atomicAdd(dst, s);
  }
  __syncthreads();
}

// ---------------------------------------------------------------------------
// Tensor Data Mover: 1-D contiguous tile load, global -> LDS.
// D# packed per CDNA5 ISA 8.3/8.4: group0 = {count=1 | lds_addr | global_addr
// | type=2}; group1 = {data_size=4B, tensor_dim0=valid elements (OOB reads
// return 0 for the tail tile), tensor_dim1=1, tile_dim0=tile_elems, 1-D}.
// Groups 2/3 zeroed (tensor rank <= 2 form). Issued by one wave; completion
// via s_wait_tensorcnt on the issuing wave, then a workgroup barrier.
// ---------------------------------------------------------------------------
__device__ __forceinline__ void tdm_load_1d(unsigned lds_off,
                                            unsigned long long gaddr,
                                            unsigned tensor_elems,
                                            unsigned tile_elems) {
  u32x4 g0;
  g0[0] = 1u;                                         // count=1, user mode
  g0[1] = lds_off;                                    // lds_addr (bytes)
  g0[2] = (unsigned)(gaddr & 0xFFFFFFFFull);          // global_addr[31:0]
  g0[3] = (unsigned)((gaddr >> 32) & 0x01FFFFFFull)   // global_addr[56:32]
          | (2u << 30);                               // type=2 ("image")
  i32x8 g1;
  g1[0] = (int)(2u << 16);                            // mask=0, data_size=4B
  g1[1] = (int)((tensor_elems & 0xFFFFu) << 16);      // tensor_dim0[15:0]
  g1[2] = (int)((tensor_elems >> 16) & 0xFFFFu)       // tensor_dim0[31:16]
          | (1 << 16);                                // tensor_dim1 = 1
  g1[3] = (int)((tile_elems & 0xFFFFu) << 16);        // tile_dim0
  g1[4] = 0;                                          // tile_dim1/2 unused
  g1[5] = (int)tensor_elems;                          // dim0_stride[31:0]
  g1[6] = 0;                                          // dim0_stride[47:32]
  g1[7] = 0;
  i32x4 z4 = {0, 0, 0, 0};
#if defined(__clang_major__) && (__clang_major__ >= 23)
  i32x8 z8 = {0, 0, 0, 0, 0, 0, 0, 0};
  __builtin_amdgcn_tensor_load_to_lds(g0, g1, z4, z4, z8, 0);
#else
  __builtin_amdgcn_tensor_load_to_lds(g0, g1, z4, z4, 0);
#endif
  __builtin_amdgcn_s_wait_tensorcnt(0);
}

// ---------------------------------------------------------------------------
// K0: zero best-prior keys and accumulators (every call; graph-replay safe).
// ---------------------------------------------------------------------------
__global__ void k_init(unsigned long long* best, float* acc, int nbm) {
  int i = blockIdx.x * blockDim.x + threadIdx.x;
  if (i < nbm) best[i] = 0ull;
  if (i < 3)   acc[i]  = 0.f;
}

// ---------------------------------------------------------------------------
// K1: per-(b,p) IoU vs all M GT boxes (GT in LDS); best-target per prior and
//     best-prior per target (packed u64 argmax, LDS atomics -> 1 global/GT).
// ---------------------------------------------------------------------------
__global__ void k_match(const float* __restrict__ dboxes,
                        const float* __restrict__ gts,
                        float* __restrict__ bt_ov, int* __restrict__ bt_idx,
                        unsigned long long* __restrict__ best,
                        int P, int M) {
  __shared__ float s_gt[256];                 // M<=64 boxes * 4
  __shared__ float s_area[64];
  __shared__ unsigned long long s_best[64];
  const int b = blockIdx.y;
  const int p = blockIdx.x * blockDim.x + threadIdx.x;
  const int tid = threadIdx.x;

  for (int i = tid; i < M * 4; i += blockDim.x)
    s_gt[i] = gts[(size_t)b * M * 4 + i];
  if (tid < M) s_best[tid] = 0ull;
  __syncthreads();
  if (tid < M) {
    float x1 = s_gt[tid*4+0], y1 = s_gt[tid*4+1];
    float x2 = s_gt[tid*4+2], y2 = s_gt[tid*4+3];
    s_area[tid] = (x2 - x1) * (y2 - y1);
  }
  __syncthreads();

  if (p < P) {
    const float d0 = dboxes[p*4+0], d1 = dboxes[p*4+1];
    const float d2 = dboxes[p*4+2], d3 = dboxes[p*4+3];
    const float a1 = (d2 - d0) * (d3 - d1);
    float bov = -1.f; int bidx = 0;
    for (int m = 0; m < M; ++m) {
      const float g0 = s_gt[m*4+0], g1 = s_gt[m*4+1];
      const float g2 = s_gt[m*4+2], g3 = s_gt[m*4+3];
      const float lx = fmaxf(d0, g0), ly = fmaxf(d1, g1);
      const float rx = fminf(d2, g2), ry = fminf(d3, g3);
      const float w = fmaxf(rx - lx, 0.f), h = fmaxf(ry - ly, 0.f);
      const float inter = w * h;
      const float iou = inter / (a1 + s_area[m] - inter);
      if (iou > bov) { bov = iou; bidx = m; }
      // iou >= 0 -> float bits order-monotonic; low word ~p gives
      // lowest-index tie-break under max.
      const unsigned long long key =
          ((unsigned long long)__float_as_uint(iou) << 32) |
          (unsigned long long)(0xFFFFFFFFu - (unsigned)p);
      atomicMax(&s_best[m], key);
    }
    bt_ov [(size_t)b * P + p] = bov;
    bt_idx[(size_t)b * P + p] = bidx;
  }
  __syncthreads();
  if (tid < M) atomicMax(&best[(size_t)b * M + tid], s_best[tid]);
}

// ---------------------------------------------------------------------------
// K2: force-match each GT to its best prior.
// ---------------------------------------------------------------------------
__global__ void k_force(const unsigned long long* __restrict__ best,
                        float* __restrict__ bt_ov, int* __restrict__ bt_idx,
                        int P, int M, int nbm) {
  int i = blockIdx.x * blockDim.x + threadIdx.x;
  if (i >= nbm) return;
  const int b = i / M, m = i % M;
  const unsigned long long key = best[i];
  const unsigned p = 0xFFFFFFFFu - (unsigned)(key & 0xFFFFFFFFull);
  bt_idx[(size_t)b * P + p] = m;
  bt_ov [(size_t)b * P + p] = 2.0f;
}

// ---------------------------------------------------------------------------
// K3: labels, encode->decode targets, decode preds, GIoU on positives,
//     accumulate loc-loss sum and positive count via WMMA block reduction.
// ---------------------------------------------------------------------------
__global__ void k_loc(const float* __restrict__ loc_preds,
                      const float* __restrict__ dboxes,
                      const float* __restrict__ gts,
                      const int* __restrict__ glabels,
                      const float* __restrict__ bt_ov,
                      const int* __restrict__ bt_idx,
                      int* __restrict__ conf_t, float* acc,
                      int P, int M) {
  __shared__ float s_gt[256];
  __shared__ int   s_lab[64];
  __shared__ float red8[8];
  const int b = blockIdx.y;
  const int p = blockIdx.x * blockDim.x + threadIdx.x;
  const int tid = threadIdx.x;

  for (int i = tid; i < M * 4; i += blockDim.x)
    s_gt[i] = gts[(size_t)b * M * 4 + i];
  if (tid < M) s_lab[tid] = glabels[(size_t)b * M + tid];
  __syncthreads();

  float gl = 0.f, cnt = 0.f;
  if (p < P) {
    const size_t bp = (size_t)b * P + p;
    const float ov = bt_ov[bp];
    const int  idx = bt_idx[bp];
    const int  lab = (ov < THRESH_C) ? 0 : s_lab[idx];
    conf_t[bp] = lab;
    if (lab > 0) {
      cnt = 1.f;
      const float d0 = dboxes[p*4+0], d1 = dboxes[p*4+1];
      const float d2 = dboxes[p*4+2], d3 = dboxes[p*4+3];
      const float dw = d2 - d0, dh = d3 - d1;
      const float dcx = d0 + dw * 0.5f, dcy = d1 + dh * 0.5f;
      const float g0 = s_gt[idx*4+0], g1 = s_gt[idx*4+1];
      const float g2 = s_gt[idx*4+2], g3 = s_gt[idx*4+3];
      const float gw = g2 - g0, gh = g3 - g1;
      const float gcx = g0 + gw * 0.5f, gcy = g1 + gh * 0.5f;
      // encode (reference eps semantics)
      const float ex = (gcx - dcx) / (dw + EPS8_C);
      const float ey = (gcy - dcy) / (dh + EPS8_C);
      const float ew = logf(gw / (dw + EPS8_C) + EPS8_C);
      const float eh = logf(gh / (dh + EPS8_C) + EPS8_C);
      // decode target
      const float tcx = ex * dw + dcx, tcy = ey * dh + dcy;
      const float tw = expf(ew) * dw,  th = expf(eh) * dh;
      const float t0 = tcx - tw*0.5f, t1 = tcy - th*0.5f;
      const float t2 = tcx + tw*0.5f, t3 = tcy + th*0.5f;
      // decode pred
      const float l0 = loc_preds[bp*4+0], l1 = loc_preds[bp*4+1];
      const float l2 = loc_preds[bp*4+2], l3 = loc_preds[bp*4+3];
      const float pcx = l0 * dw + dcx, pcy = l1 * dh + dcy;
      const float pw = expf(l2) * dw,  ph = expf(l3) * dh;
      const float q0 = pcx - pw*0.5f, q1 = pcy - ph*0.5f;
      const float q2 = pcx + pw*0.5f, q3 = pcy + ph*0.5f;
      // GIoU
      const float ix = fmaxf(q0, t0), iy = fmaxf(q1, t1);
      const float ax = fminf(q2, t2), ay = fminf(q3, t3);
      const float iw = fmaxf(ax - ix, 0.f), ih = fmaxf(ay - iy, 0.f);
      const float inter = iw * ih;
      const float pa = (q2 - q0) * (q3 - q1);
      const float ta = (t2 - t0) * (t3 - t1);
      const float uni = pa + ta - inter;
      const float iou = inter / (uni + EPS7_C);
      const float e0 = fminf(q0, t0), e1 = fminf(q1, t1);
      const float e2 = fmaxf(q2, t2), e3 = fmaxf(q3, t3);
      const float ew2 = fmaxf(e2 - e0, 0.f), eh2 = fmaxf(e3 - e1, 0.f);
      const float encl = ew2 * eh2;
      const float giou = iou - (encl - uni) / (encl + EPS7_C);
      gl = 1.f - giou;
    }
  }
  block_accum(gl,  &acc[0], red8);
  block_accum(cnt, &acc[2], red8);
}

// ---------------------------------------------------------------------------
// K4: focal loss over all anchors (the 132 MB streaming pass).
//     A TDM tensor_load_to_lds stages the 256-anchor x C tile into LDS (one
//     bulk DMA per block, TENSORcnt-tracked), then threads read their row
//     from LDS (row stride C=21 odd -> conflict-free). Unrolled over C,
//     dynamic class gather via select chain (no spill).
// ---------------------------------------------------------------------------
template <int CC>
__global__ void k_focal_t(const float* __restrict__ conf,
                          const int* __restrict__ conf_t,
                          float* acc, int N) {
  __shared__ float s_x[256 * CC];
  __shared__ float red8[8];
  const int tid  = threadIdx.x;
  const int base = blockIdx.x * 256;
  const int i    = base + tid;

  // Wave 0 issues the tile DMA and waits on TENSORcnt; barrier releases all.
  if ((tid >> 5) == 0) {
    const unsigned tile_elems = 256u * (unsigned)CC;
    size_t rem = ((size_t)N - (size_t)base) * (unsigned)CC;
    const unsigned tensor_elems =
        (rem < (size_t)tile_elems) ? (unsigned)rem : tile_elems;
    const unsigned lds_off =
        (unsigned)(unsigned long long)(uintptr_t)(&s_x[0]);
    const unsigned long long gaddr =
        (unsigned long long)(uintptr_t)conf +
        (unsigned long long)base * (unsigned)CC * 4ull;
    tdm_load_1d(lds_off, gaddr, tensor_elems, tile_elems);
  }
  __syncthreads();

  float fl = 0.f;
  if (i < N) {
    const float* x = &s_x[tid * CC];
    const int t = conf_t[i];
    float xs[CC];
    float m = -3.402823466e38f;
#pragma unroll
    for (int c = 0; c < CC; ++c) { xs[c] = x[c]; m = fmaxf(m, xs[c]); }
    float s = 0.f;
    float xt = 0.f;
#pragma unroll
    for (int c = 0; c < CC; ++c) {
      s += expf(xs[c] - m);
      xt = (c == t) ? xs[c] : xt;
    }
    const float logp = xt - m - logf(s);
    const float ce = -logp;
    const float pt = expf(logp);
    const float om = fmaxf(1.f - pt, 0.f);
    fl = ALPHA_C * om * sqrtf(om) * ce;           // (1-pt)^1.5
  }
  block_accum(fl, &acc[1], red8);
}

// Generic-C fallback (plain streaming loads, no register array).
__global__ void k_focal_gen(const float* __restrict__ conf,
                            const int* __restrict__ conf_t,
                            float* acc, int N, int C) {
  __shared__ float red8[8];
  const int i = blockIdx.x * blockDim.x + threadIdx.x;
  float fl = 0.f;
  if (i < N) {
    const float* x = conf + (size_t)i * C;
    const int t = conf_t[i];
    float m = -3.402823466e38f;
    for (int c = 0; c < C; ++c) m = fmaxf(m, x[c]);
    float s = 0.f, xt = 0.f;
    for (int c = 0; c < C; ++c) {
      const float v = x[c];
      s += expf(v - m);
      xt = (c == t) ? v : xt;
    }
    const float logp = xt - m - logf(s);
    const float ce = -logp;
    const float pt = expf(logp);
    const float om = fmaxf(1.f - pt, 0.f);
    fl = ALPHA_C * om * sqrtf(om) * ce;
  }
  block_accum(fl, &acc[1], red8);
}

// ---------------------------------------------------------------------------
// K5: finalize scalar.
// ---------------------------------------------------------------------------
__global__ void k_final(const float* __restrict__ acc, float* __restrict__ out) {
  const float np = acc[2];
  const float denom = fmaxf(np, 1.f);
  const float total = (acc[0] + acc[1]) / denom;
  out[0] = (np == 0.f) ? 0.f : total;
}

// ---------------------------------------------------------------------------
extern "C" void kernel_launch(void* const* d_in, const int* in_sizes, int n_in,
                              void* d_out, int out_size, void* d_ws,
                              size_t ws_size, hipStream_t stream) {
  const float* loc_preds  = (const float*)d_in[0];
  const float* conf_preds = (const float*)d_in[1];
  const float* dboxes     = (const float*)d_in[2];
  const float* gts        = (const float*)d_in[3];
  const int*   glabels    = (const int*)d_in[4];

  const int P = in_sizes[2] / 4;
  const int B = in_sizes[0] / (4 * P);
  const int M = in_sizes[4] / B;
  const int C = in_sizes[1] / (in_sizes[0] / 4);
  const size_t BP = (size_t)B * P;
  const int nbm = B * M;

  // Workspace carve-out (256B aligned slices).
  char* ws = (char*)d_ws;
  auto carve = [&](size_t bytes) {
    char* p = ws;
    ws += (bytes + 255) & ~(size_t)255;
    return p;
  };
  float* bt_ov  = (float*)carve(BP * sizeof(float));
  int*   bt_idx = (int*)carve(BP * sizeof(int));
  int*   conf_t = (int*)carve(BP * sizeof(int));
  unsigned long long* best = (unsigned long long*)carve(nbm * sizeof(unsigned long long));
  float* acc    = (float*)carve(3 * sizeof(float));

  const int TB = 256;
  k_init<<<(nbm + TB - 1) / TB, TB, 0, stream>>>(best, acc, nbm);

  dim3 gmatch((P + TB - 1) / TB, B);
  k_match<<<gmatch, TB, 0, stream>>>(dboxes, gts, bt_ov, bt_idx, best, P, M);

  k_force<<<(nbm + TB - 1) / TB, TB, 0, stream>>>(best, bt_ov, bt_idx, P, M, nbm);

  k_loc<<<gmatch, TB, 0, stream>>>(loc_preds, dboxes, gts, glabels, bt_ov,
                                   bt_idx, conf_t, acc, P, M);

  const int N = (int)BP;
  const int gN = (N + TB - 1) / TB;
  if (C == 21)
    k_focal_t<21><<<gN, TB, 0, stream>>>(conf_preds, conf_t, acc, N);
  else
    k_focal_gen<<<gN, TB, 0, stream>>>(conf_preds, conf_t, acc, N, C);

  k_final<<<1, 1, 0, stream>>>(acc, (float*)d_out);
}